// CRF_8237747274169
// MI455X (gfx1250) — compile-verified
//
#include <hip/hip_runtime.h>

typedef __attribute__((ext_vector_type(2))) float v2f;
typedef __attribute__((ext_vector_type(4))) float v4f;
typedef __attribute__((ext_vector_type(8))) float v8f;

#define BSZ   32
#define TLEN  512
#define LSZ   64
#define FST   (TLEN * LSZ)   // 32768 floats per batch in feats
#define LN2   0.6931471805599453f

// Workgroup barrier that only drains LDS (dscnt). Safe here: all cross-thread
// data lives in LDS; global traffic is read-only loads into private VGPRs.
// Crucially this does NOT flush loadcnt, so prefetched global loads stay in
// flight across the barrier and are waited only at their use.
__device__ __forceinline__ void lds_barrier() {
  asm volatile("s_wait_dscnt 0x0\n\t"
               "s_barrier_signal -1\n\t"
               "s_barrier_wait -1" ::: "memory");
}

// ---------------------------------------------------------------------------
// Forward (log-partition) scan: one workgroup, 8 waves, each wave owns one
// 16x16 tile of D(64 x 32) = exp(T)(64x64) @ V(64x32) via v_wmma_f32_16x16x4_f32.
// fv is kept in a per-batch max-shifted (relative) domain; the cumulative
// shift is carried in registers by the reduction-layout lanes.
// ---------------------------------------------------------------------------
__launch_bounds__(256, 1)
__global__ void crf_forward_kernel(const float* __restrict__ feats,
                                   const float* __restrict__ T,
                                   float* __restrict__ fs_out) {
  __shared__ float eT[64][65];     // exp(T), padded
  __shared__ float fv[64][33];     // state (relative domain): fv[j][b], padded
  __shared__ float Vx[64][33];     // exp(fv - m), padded

  const int tid  = threadIdx.x;
  const int lane = tid & 31;
  const int wave = tid >> 5;       // 0..7
  const int half = lane >> 4;      // 0 or 1
  const int l15  = lane & 15;
  const int r0   = (wave >> 1) * 16;   // WMMA row tile origin (i)
  const int ct0  = (wave & 1) * 16;    // WMMA col tile origin (batch)

  // Reduction-phase ownership: batch bred, 8 states per lane
  const int bred = wave * 4 + (lane & 3);
  const int jrow = lane >> 2;          // 0..7 -> j = jrow*8 + q

  // One-time: exp(T) into LDS
  for (int idx = tid; idx < 64 * 64; idx += 256)
    eT[idx >> 6][idx & 63] = __expf(T[idx]);

  // Init fv = feats[:, 0, :]   (fv[j][b])
  for (int q = 0; q < 8; ++q) {
    int j = jrow * 8 + q;
    fv[j][bred] = feats[bred * FST + j];
  }
  __syncthreads();

  // Hoist constant A fragments: expT rows of this wave's tile, K=64 in 16 chunks
  // A 16x4 f32 layout: lanes 0-15 -> K = kc*4 + {0,1}; lanes 16-31 -> K = kc*4 + {2,3}
  v2f afrag[16];
  {
    const int m = r0 + l15;
    for (int kc = 0; kc < 16; ++kc) {
      const int k = kc * 4 + half * 2;
      v2f a; a.x = eT[m][k]; a.y = eT[m][k + 1];
      afrag[kc] = a;
    }
  }

  const int bcol = ct0 + l15;      // output batch column of this lane
  float off = 0.f;                 // cumulative max-shift for batch bred

  for (int t = 1; t < TLEN; ++t) {
    // Issue this step's feats loads NOW; with the dscnt-only barrier they
    // remain in flight across the barrier and under the whole WMMA chain.
    const float* fb = feats + bcol * FST + t * LSZ + r0 + half * 8;
    v4f f0 = *(const v4f*)fb;
    v4f f1 = *(const v4f*)(fb + 4);

    // ---- per-batch max via in-wave shuffle reduction ----
    float fvl[8];
    for (int q = 0; q < 8; ++q) fvl[q] = fv[jrow * 8 + q][bred];
    float mx = fvl[0];
    for (int q = 1; q < 8; ++q) mx = fmaxf(mx, fvl[q]);
    mx = fmaxf(mx, __shfl_xor(mx, 4));
    mx = fmaxf(mx, __shfl_xor(mx, 8));
    mx = fmaxf(mx, __shfl_xor(mx, 16));
    off += mx;                     // carry the shift in a register

    // ---- V = exp(fv - m) ----
    for (int q = 0; q < 8; ++q)
      Vx[jrow * 8 + q][bred] = __expf(fvl[q] - mx);
    lds_barrier();

    // ---- D = expT @ V via 16 f32 WMMAs, 4 independent accumulator chains ----
    v8f acc0 = {0.f,0.f,0.f,0.f,0.f,0.f,0.f,0.f};
    v8f acc1 = {0.f,0.f,0.f,0.f,0.f,0.f,0.f,0.f};
    v8f acc2 = {0.f,0.f,0.f,0.f,0.f,0.f,0.f,0.f};
    v8f acc3 = {0.f,0.f,0.f,0.f,0.f,0.f,0.f,0.f};
    for (int kc = 0; kc < 16; kc += 4) {
      {
        const int k = (kc + 0) * 4 + half * 2;
        v2f b; b.x = Vx[k][bcol]; b.y = Vx[k + 1][bcol];
        acc0 = __builtin_amdgcn_wmma_f32_16x16x4_f32(false, afrag[kc + 0], false, b,
                                                     (short)0, acc0, false, false);
      }
      {
        const int k = (kc + 1) * 4 + half * 2;
        v2f b; b.x = Vx[k][bcol]; b.y = Vx[k + 1][bcol];
        acc1 = __builtin_amdgcn_wmma_f32_16x16x4_f32(false, afrag[kc + 1], false, b,
                                                     (short)0, acc1, false, false);
      }
      {
        const int k = (kc + 2) * 4 + half * 2;
        v2f b; b.x = Vx[k][bcol]; b.y = Vx[k + 1][bcol];
        acc2 = __builtin_amdgcn_wmma_f32_16x16x4_f32(false, afrag[kc + 2], false, b,
                                                     (short)0, acc2, false, false);
      }
      {
        const int k = (kc + 3) * 4 + half * 2;
        v2f b; b.x = Vx[k][bcol]; b.y = Vx[k + 1][bcol];
        acc3 = __builtin_amdgcn_wmma_f32_16x16x4_f32(false, afrag[kc + 3], false, b,
                                                     (short)0, acc3, false, false);
      }
    }
    v8f c = (acc0 + acc1) + (acc2 + acc3);

    // ---- fv_rel_new[i][b] = feats[b,t,i] + log(D[i][b])  (no +m: shift is
    // carried in `off`; max-shift cancels identically step to step) ----
    // C/D layout: component r -> M = r0 + half*8 + r, N = bcol.
    // Raw v_log_f32 is safe: D is a sum of 64 positive normal-range terms.
    for (int r = 0; r < 8; ++r) {
      const int i = r0 + half * 8 + r;
      const float fval = (r < 4) ? f0[r] : f1[r - 4];
      fv[i][bcol] = fval + __builtin_amdgcn_logf(c[r]) * LN2;
    }
    lds_barrier();
  }

  // ---- final forward_score[b] = off[b] + logsumexp_j fv_rel[j][b] ----
  float fvl[8];
  for (int q = 0; q < 8; ++q) fvl[q] = fv[jrow * 8 + q][bred];
  float mx = fvl[0];
  for (int q = 1; q < 8; ++q) mx = fmaxf(mx, fvl[q]);
  mx = fmaxf(mx, __shfl_xor(mx, 4));
  mx = fmaxf(mx, __shfl_xor(mx, 8));
  mx = fmaxf(mx, __shfl_xor(mx, 16));
  float s = 0.f;
  for (int q = 0; q < 8; ++q) s += __expf(fvl[q] - mx);
  s += __shfl_xor(s, 4);
  s += __shfl_xor(s, 8);
  s += __shfl_xor(s, 16);
  if (jrow == 0) fs_out[bred] = off + mx + __builtin_amdgcn_logf(s) * LN2;
}

// ---------------------------------------------------------------------------
// Gold score + final output: out[b] = forward_score[b] - (emit[b] + trans[b])
// ---------------------------------------------------------------------------
__launch_bounds__(256, 1)
__global__ void crf_gold_kernel(const float* __restrict__ feats,
                                const long long* __restrict__ tags,
                                const float* __restrict__ T,
                                const float* __restrict__ fs,
                                float* __restrict__ out) {
  __shared__ float red[256];
  const int b = blockIdx.x;
  const int tid = threadIdx.x;
  float acc = 0.f;
  for (int t = tid; t < TLEN; t += 256) {
    const int tg = ((int)tags[b * TLEN + t]) & 63;
    acc += feats[b * FST + t * LSZ + tg];
    if (t >= 1) {
      const int tp = ((int)tags[b * TLEN + t - 1]) & 63;
      acc += T[tg * LSZ + tp];
    }
  }
  red[tid] = acc;
  __syncthreads();
  for (int s = 128; s > 0; s >>= 1) {
    if (tid < s) red[tid] += red[tid + s];
    __syncthreads();
  }
  if (tid == 0) out[b] = fs[b] - red[0];
}

extern "C" void kernel_launch(void* const* d_in, const int* in_sizes, int n_in,
                              void* d_out, int out_size, void* d_ws, size_t ws_size,
                              hipStream_t stream) {
  (void)in_sizes; (void)n_in; (void)out_size; (void)ws_size;
  const float*     feats = (const float*)d_in[0];
  const long long* tags  = (const long long*)d_in[1];
  const float*     T     = (const float*)d_in[2];
  float* out = (float*)d_out;
  float* fs  = (float*)d_ws;   // 32 floats of scratch: forward scores

  crf_forward_kernel<<<1, 256, 0, stream>>>(feats, T, fs);
  crf_gold_kernel<<<BSZ, 256, 0, stream>>>(feats, tags, T, fs, out);
}